// DenseProduct_28698971471972
// MI455X (gfx1250) — compile-verified
//
#include <hip/hip_runtime.h>

// CDNA5 / gfx1250, wave32.
// out[s,d,b, i*32+j] = x[2s,d,b,i] + x[2s+1,d,b,j]   (S=64,D=16,B=64,N=32)
//
// Outer-sum via exact f32 rank-2 WMMA:
//   A(16x4) = [a_i, 1, 0, 0], B(4x16) = [1; b_j; 0; 0] -> D = a_i + b_j
// One wave32 per (s,d,b) tile: 4x V_WMMA_F32_16X16X4_F32 -> 32x32 f32 tile.
// Output (256 MB) > L2 (192 MB) and never re-read -> non-temporal stores.

typedef __attribute__((ext_vector_type(2))) float v2f;
typedef __attribute__((ext_vector_type(8))) float v8f;

__global__ __launch_bounds__(256) void DenseProduct_outer_sum_wmma(
    const float* __restrict__ x, float* __restrict__ out, int nTiles) {
  const int gid  = blockIdx.x * 256 + (int)threadIdx.x;
  int wave       = gid >> 5;
  if (wave >= nTiles) wave = nTiles - 1;  // grid divides exactly; never taken
  const int lane = threadIdx.x & 31;
  const int l    = lane & 15;
  const bool lo  = lane < 16;  // lanes 0-15 carry K=0/1; lanes 16-31 (K=2/3) = 0

  // wave = (s*16 + d)*64 + b
  const int b  = wave & 63;
  const int sd = wave >> 6;      // s*16 + d
  const int s  = sd >> 4;
  const int d  = sd & 15;

  // input rows: x[2s,d,b,:] and x[2s+1,d,b,:]; scope stride = D*B*N = 32768
  const float* __restrict__ arow =
      x + ((((size_t)(2 * s) * 16 + d) * 64 + b) * 32);
  const float* __restrict__ brow = arow + (size_t)16 * 64 * 32;

  // broadcast loads (16 distinct addrs per wave, L1-resident, RT policy)
  const float a0 = arow[l];
  const float a1 = arow[l + 16];
  const float b0 = brow[l];
  const float b1 = brow[l + 16];

  float* __restrict__ o = out + (size_t)wave * 1024;
  const int m0 = lo ? 0 : 8;  // D layout: VGPR r holds row r (lanes 0-15) / r+8

#pragma unroll
  for (int ti = 0; ti < 2; ++ti) {
#pragma unroll
    for (int tj = 0; tj < 2; ++tj) {
      const float av = ti ? a1 : a0;
      const float bv = tj ? b1 : b0;

      // A 16x4 f32: per-lane VGPR0 = K0, VGPR1 = K1 (lanes 0-15), K2/K3 zero
      v2f A;
      A.x = lo ? av : 0.0f;    // A[i][0] = a_i
      A.y = lo ? 1.0f : 0.0f;  // A[i][1] = 1
      // B 4x16 f32: per-lane VGPR0 = row K0, VGPR1 = row K1
      v2f Bm;
      Bm.x = lo ? 1.0f : 0.0f;  // B[0][j] = 1
      Bm.y = lo ? bv : 0.0f;    // B[1][j] = b_j
      v8f C = {};

      // D = A x B + 0 : exact f32, one rounding (RNE), matches v_add_f32
      v8f D2 = __builtin_amdgcn_wmma_f32_16x16x4_f32(
          /*neg_a=*/false, A, /*neg_b=*/false, Bm,
          /*c_mod=*/(short)0, C, /*reuse_a=*/false, /*reuse_b=*/false);

      float* __restrict__ ob = o + (size_t)(ti * 16) * 32 + tj * 16;
#pragma unroll
      for (int r = 0; r < 8; ++r) {
        // row = ti*16 + m0 + r, col = tj*16 + l; streaming store, bypass L2 fill
        __builtin_nontemporal_store(D2[r], &ob[(m0 + r) * 32 + l]);
      }
    }
  }
}

extern "C" void kernel_launch(void* const* d_in, const int* in_sizes, int n_in,
                              void* d_out, int out_size, void* d_ws, size_t ws_size,
                              hipStream_t stream) {
  (void)in_sizes; (void)n_in; (void)d_ws; (void)ws_size;
  const float* x = (const float*)d_in[0];
  float* out     = (float*)d_out;

  const int nTiles  = out_size >> 10;            // (s,d,b) tiles, 1024 outputs each
  const int threads = nTiles * 32;               // one wave32 per tile
  const int blocks  = (threads + 255) / 256;     // 8 waves per 256-thread block
  DenseProduct_outer_sum_wmma<<<blocks, 256, 0, stream>>>(x, out, nTiles);
}